// GINet_4836133175446
// MI455X (gfx1250) — compile-verified
//
#include <hip/hip_runtime.h>

typedef __attribute__((ext_vector_type(2))) float v2f;
typedef __attribute__((ext_vector_type(8))) float v8f;

#define N_NODES 50000
#define N_EDGES 1600000
#define N_GRAPHS 512

// ---------------- zero fill (float4) ----------------
__global__ void k_zero(float* __restrict__ p, int n4) {
  int i = blockIdx.x * blockDim.x + threadIdx.x;
  if (i < n4) ((float4*)p)[i] = make_float4(0.f, 0.f, 0.f, 0.f);
}

// ---------------- sparse aggregation: out[row[e]] += X[col[e]] ----------------
// C = feature width (multiple of 4). One thread per (edge, float4-chunk).
template<int C>
__global__ void k_agg(const float* __restrict__ X, const int* __restrict__ rowi,
                      const int* __restrict__ coli, float* __restrict__ out, int nE) {
  constexpr int CH = C / 4;
  int t = blockIdx.x * blockDim.x + threadIdx.x;
  if (t >= nE * CH) return;
  int e  = t / CH;
  int ch = t % CH;
  int r = rowi[e], c = coli[e];
  float4 v = ((const float4*)(X + (size_t)c * C))[ch];
  float* dst = out + (size_t)r * C + ch * 4;
  atomicAdd(dst + 0, v.x);
  atomicAdd(dst + 1, v.y);
  atomicAdd(dst + 2, v.z);
  atomicAdd(dst + 3, v.w);
}

// ---------------- WMMA GEMM: Y[M,COUT] = act(X[M,CIN] @ W^T + bias) ----------------
// W row-major [COUT, CIN]. One wave per 16-row tile. V_WMMA_F32_16X16X4_F32 over K.
// A frag (16x4 f32): lane L -> M = L&15, VGPR0/1 = K = 2*(L>>4), 2*(L>>4)+1.
// B frag (4x16 f32): lane L -> N = L&15, same K split.
// C/D (16x16 f32, 8 VGPRs): VGPR v -> M = v + 8*(L>>4), N = L&15.
template<int CIN, int COUT>
__global__ void k_gemm_wmma(const float* __restrict__ X, const float* __restrict__ W,
                            const float* __restrict__ bias, float* __restrict__ Y,
                            int ntiles, int do_relu) {
  int lane = threadIdx.x & 31;
  int wave = threadIdx.x >> 5;
  int tile = blockIdx.x * (blockDim.x >> 5) + wave;
  if (tile >= ntiles) return;                 // wave-uniform: EXEC stays all-ones
  const int m  = lane & 15;
  const int hi = lane >> 4;

  const float* Xrow = X + (size_t)(tile * 16 + m) * CIN;
  v2f af[CIN / 4];
#pragma unroll
  for (int kb = 0; kb < CIN / 4; ++kb) {
    int k = kb * 4 + 2 * hi;
    af[kb].x = Xrow[k];
    af[kb].y = Xrow[k + 1];
  }

#pragma unroll
  for (int nt = 0; nt < COUT / 16; ++nt) {
    int n = m + nt * 16;
    const float* Wrow = W + (size_t)n * CIN;
    v8f acc = {};
#pragma unroll
    for (int kb = 0; kb < CIN / 4; ++kb) {
      int k = kb * 4 + 2 * hi;
      v2f b;
      b.x = Wrow[k];
      b.y = Wrow[k + 1];
      acc = __builtin_amdgcn_wmma_f32_16x16x4_f32(
          /*neg_a=*/false, af[kb], /*neg_b=*/false, b,
          /*c_mod=*/(short)0, acc, /*reuse_a=*/false, /*reuse_b=*/false);
    }
    float bval = bias ? bias[n] : 0.f;        // bias indexed by this lane's N column
    float* Yt = Y + (size_t)tile * 16 * COUT + nt * 16;
#pragma unroll
    for (int v = 0; v < 8; ++v) {
      int mr = v + 8 * hi;
      float val = acc[v] + bval;
      if (do_relu) val = fmaxf(val, 0.f);
      Yt[(size_t)mr * COUT + m] = val;
    }
  }
}

// ---------------- graph pooling ----------------
__global__ void k_count(const int* __restrict__ batch, float* __restrict__ cnt, int n) {
  int i = blockIdx.x * blockDim.x + threadIdx.x;
  if (i < n) atomicAdd(&cnt[batch[i]], 1.0f);
}

__global__ void k_pool(const float* __restrict__ h2, const float* __restrict__ he2,
                       const int* __restrict__ batch, float* __restrict__ P, int n) {
  int t = blockIdx.x * blockDim.x + threadIdx.x;
  if (t >= n * 64) return;
  int node = t >> 6, f = t & 63;
  float v = (f < 32) ? h2[(size_t)node * 32 + f] : he2[(size_t)node * 32 + (f - 32)];
  atomicAdd(&P[(size_t)batch[node] * 64 + f], v);
}

__global__ void k_pooldiv(float* __restrict__ P, const float* __restrict__ cnt) {
  int t = blockIdx.x * blockDim.x + threadIdx.x;
  if (t >= N_GRAPHS * 64) return;
  P[t] /= fmaxf(cnt[t >> 6], 1.0f);
}

// ---------------- fc2: [512,128] @ [128,1] + b ----------------
__global__ void k_fc2(const float* __restrict__ g1, const float* __restrict__ w,
                      const float* __restrict__ b, float* __restrict__ out) {
  int g = blockIdx.x * blockDim.x + threadIdx.x;
  if (g >= N_GRAPHS) return;
  float s = b[0];
#pragma unroll 8
  for (int k = 0; k < 128; ++k) s += g1[(size_t)g * 128 + k] * w[k];
  out[g] = s;
}

extern "C" void kernel_launch(void* const* d_in, const int* in_sizes, int n_in,
                              void* d_out, int out_size, void* d_ws, size_t ws_size,
                              hipStream_t stream) {
  const float* x       = (const float*)d_in[0];
  const int*   eidx    = (const int*)d_in[1];
  const int*   batch   = (const int*)d_in[3];
  const float* c1_fc   = (const float*)d_in[4];
  const float* c2_fc   = (const float*)d_in[7];
  const float* c1e_fc  = (const float*)d_in[10];
  const float* c2e_fc  = (const float*)d_in[13];
  const float* fc1_w   = (const float*)d_in[16];
  const float* fc1_b   = (const float*)d_in[17];
  const float* fc2_w   = (const float*)d_in[18];
  const float* fc2_b   = (const float*)d_in[19];
  const int* row = eidx;
  const int* col = eidx + N_EDGES;

  // workspace layout (floats)
  float* ws   = (float*)d_ws;
  float* aggx = ws;                              // N*64
  float* h1   = aggx + (size_t)N_NODES * 64;     // N*16
  float* he1  = h1   + (size_t)N_NODES * 16;     // N*16
  float* aggb = he1  + (size_t)N_NODES * 16;     // N*16 (reused)
  float* h2   = aggb + (size_t)N_NODES * 16;     // N*32
  float* he2  = h2   + (size_t)N_NODES * 32;     // N*32
  float* P    = he2  + (size_t)N_NODES * 32;     // G*64
  float* cnt  = P    + (size_t)N_GRAPHS * 64;    // G
  float* g1   = cnt  + (size_t)N_GRAPHS;         // G*128

  const int TPB = 256;
  const int ntiles = N_NODES / 16;               // 3125 exactly

  auto nblk = [](int total, int tpb) { return (total + tpb - 1) / tpb; };

  // layer 1: agg(x) shared by both branches
  k_zero<<<nblk(N_NODES * 64 / 4, TPB), TPB, 0, stream>>>(aggx, N_NODES * 64 / 4);
  k_agg<64><<<nblk(N_EDGES * 16, TPB), TPB, 0, stream>>>(x, row, col, aggx, N_EDGES);
  k_gemm_wmma<64, 16><<<nblk(ntiles, 8), TPB, 0, stream>>>(aggx, c1_fc,  nullptr, h1,  ntiles, 1);
  k_gemm_wmma<64, 16><<<nblk(ntiles, 8), TPB, 0, stream>>>(aggx, c1e_fc, nullptr, he1, ntiles, 1);

  // layer 2, branch h
  k_zero<<<nblk(N_NODES * 16 / 4, TPB), TPB, 0, stream>>>(aggb, N_NODES * 16 / 4);
  k_agg<16><<<nblk(N_EDGES * 4, TPB), TPB, 0, stream>>>(h1, row, col, aggb, N_EDGES);
  k_gemm_wmma<16, 32><<<nblk(ntiles, 8), TPB, 0, stream>>>(aggb, c2_fc, nullptr, h2, ntiles, 1);

  // layer 2, branch he
  k_zero<<<nblk(N_NODES * 16 / 4, TPB), TPB, 0, stream>>>(aggb, N_NODES * 16 / 4);
  k_agg<16><<<nblk(N_EDGES * 4, TPB), TPB, 0, stream>>>(he1, row, col, aggb, N_EDGES);
  k_gemm_wmma<16, 32><<<nblk(ntiles, 8), TPB, 0, stream>>>(aggb, c2e_fc, nullptr, he2, ntiles, 1);

  // pooling (scatter-mean over graphs)
  k_zero<<<nblk(N_GRAPHS * 64 / 4, TPB), TPB, 0, stream>>>(P, N_GRAPHS * 64 / 4);
  k_zero<<<nblk(N_GRAPHS / 4, TPB), TPB, 0, stream>>>(cnt, N_GRAPHS / 4);
  k_count<<<nblk(N_NODES, TPB), TPB, 0, stream>>>(batch, cnt, N_NODES);
  k_pool<<<nblk(N_NODES * 64, TPB), TPB, 0, stream>>>(h2, he2, batch, P, N_NODES);
  k_pooldiv<<<nblk(N_GRAPHS * 64, TPB), TPB, 0, stream>>>(P, cnt);

  // head MLP
  k_gemm_wmma<64, 128><<<nblk(N_GRAPHS / 16, 8), TPB, 0, stream>>>(P, fc1_w, fc1_b, g1, N_GRAPHS / 16, 1);
  k_fc2<<<nblk(N_GRAPHS, TPB), TPB, 0, stream>>>(g1, fc2_w, fc2_b, (float*)d_out);
}